// Skipgram_12738873000106
// MI455X (gfx1250) — compile-verified
//
#include <hip/hip_runtime.h>

typedef __attribute__((ext_vector_type(2))) float v2f;
typedef __attribute__((ext_vector_type(8))) float v8f;

#define DIM    128   // embedding dim
#define NEG_K  20    // negatives per row
#define TILE   16    // batch rows per WMMA tile (one wave each)
#define WPB    2     // waves per block (48KB LDS / block)

__device__ __forceinline__ float log_sigmoid(float x) {
    // stable: logσ(x) = min(x,0) - log1p(exp(-|x|))
    float t = __builtin_fabsf(x);
    float l = -log1pf(__expf(-t));
    return (x < 0.0f) ? (x + l) : l;
}

__device__ __forceinline__ float sel8(v8f c, int i) {
    float r = c[0];
    r = (i == 1) ? c[1] : r;
    r = (i == 2) ? c[2] : r;
    r = (i == 3) ? c[3] : r;
    r = (i == 4) ? c[4] : r;
    r = (i == 5) ? c[5] : r;
    r = (i == 6) ? c[6] : r;
    r = (i == 7) ? c[7] : r;
    return r;
}

__global__ __launch_bounds__(WPB * 32)
void skipgram_tile_kernel(const float* __restrict__ u_emb,
                          const float* __restrict__ v_emb,
                          const int*   __restrict__ u_pos,
                          const int*   __restrict__ v_pos,
                          const int*   __restrict__ v_neg,
                          float*       __restrict__ partials,
                          int n_tiles)
{
    __shared__ float ldsU[WPB][TILE][DIM];   // gathered u rows
    __shared__ float ldsP[WPB][TILE][DIM];   // gathered v_pos rows
    __shared__ float ldsN[WPB][TILE][DIM];   // sum over K gathered v_neg rows

    const int lane = threadIdx.x & 31;
    const int wv   = threadIdx.x >> 5;
    const int tile = blockIdx.x * WPB + wv;
    if (tile >= n_tiles) return;             // wave-uniform branch
    const int row0 = tile * TILE;
    const int col  = lane * 4;               // 32 lanes x float4 = one 128-f row

    // ---- gather phase: coalesced 512B row reads, reduce negatives on the fly
    for (int i = 0; i < TILE; ++i) {
        const int  r  = row0 + i;
        const long up = u_pos[r];
        const long vp = v_pos[r];
        float4 u4 = *(const float4*)(u_emb + up * DIM + col);
        float4 p4 = *(const float4*)(v_emb + vp * DIM + col);
        float4 acc = make_float4(0.f, 0.f, 0.f, 0.f);
        const int* vn = v_neg + (long)r * NEG_K;
        #pragma unroll
        for (int k = 0; k < NEG_K; ++k) {
            const float4 nv = *(const float4*)(v_emb + (long)vn[k] * DIM + col);
            acc.x += nv.x; acc.y += nv.y; acc.z += nv.z; acc.w += nv.w;
        }
        *(float4*)&ldsU[wv][i][col] = u4;
        *(float4*)&ldsP[wv][i][col] = p4;
        *(float4*)&ldsN[wv][i][col] = acc;
    }
    asm volatile("s_wait_dscnt 0" ::: "memory");  // LDS writes visible to whole wave

    // ---- WMMA phase: c1 = U*Vpos^T, c2 = U*(SumVneg)^T  (f32, 16x16x4, K-chunks)
    // A(16x4) lane layout: row = lane&15, K = kc + 2*(lane>=16) .. +1
    // B(4x16) from row-major V^T has the identical per-lane fetch pattern.
    const int rsel = lane & 15;
    const int koff = (lane >> 4) << 1;       // 0 for lanes 0-15, 2 for lanes 16-31
    v8f c1 = {};
    v8f c2 = {};
    #pragma unroll 4
    for (int kc = 0; kc < DIM; kc += 4) {
        v2f a  = *(const v2f*)&ldsU[wv][rsel][kc + koff];
        v2f b1 = *(const v2f*)&ldsP[wv][rsel][kc + koff];
        v2f b2 = *(const v2f*)&ldsN[wv][rsel][kc + koff];
        c1 = __builtin_amdgcn_wmma_f32_16x16x4_f32(false, a, false, b1,
                                                   (short)0, c1, false, false);
        c2 = __builtin_amdgcn_wmma_f32_16x16x4_f32(false, a, false, b2,
                                                   (short)0, c2, false, false);
    }

    // ---- diagonal extraction: (m,m) is lane m vgpr m (m<8), lane m+16 vgpr m-8 (m>=8)
    float contrib = 0.f;
    const bool active = (lane < 8) || (lane >= 24);
    const int  ci     = (lane < 8) ? lane : (lane - 24);
    if (active) {
        float s1 = sel8(c1, ci);
        float s2 = sel8(c2, ci);
        contrib = log_sigmoid(s1) + log_sigmoid(-s2);
    }
    #pragma unroll
    for (int m = 16; m > 0; m >>= 1)
        contrib += __shfl_xor(contrib, m, 32);
    if (lane == 0) partials[tile] = contrib;
}

__global__ __launch_bounds__(256)
void skipgram_reduce_kernel(const float* __restrict__ partials,
                            float* __restrict__ out, int n, float scale)
{
    __shared__ float s[256];
    float acc = 0.f;
    for (int i = threadIdx.x; i < n; i += 256) acc += partials[i];
    s[threadIdx.x] = acc;
    __syncthreads();
    for (int off = 128; off > 0; off >>= 1) {
        if ((int)threadIdx.x < off) s[threadIdx.x] += s[threadIdx.x + off];
        __syncthreads();
    }
    if (threadIdx.x == 0) out[0] = -s[0] * scale;   // loss = -sum / B
}

extern "C" void kernel_launch(void* const* d_in, const int* in_sizes, int n_in,
                              void* d_out, int out_size, void* d_ws, size_t ws_size,
                              hipStream_t stream)
{
    const float* u_emb = (const float*)d_in[0];
    const float* v_emb = (const float*)d_in[1];
    const int*   u_pos = (const int*)d_in[2];
    const int*   v_pos = (const int*)d_in[3];
    const int*   v_neg = (const int*)d_in[4];

    const int B       = in_sizes[2];          // 16384
    const int n_tiles = B / TILE;             // 1024 (B is a multiple of 16)
    float* partials   = (float*)d_ws;         // n_tiles * 4 bytes of scratch

    const int blocks = (n_tiles + WPB - 1) / WPB;
    skipgram_tile_kernel<<<blocks, WPB * 32, 0, stream>>>(
        u_emb, v_emb, u_pos, v_pos, v_neg, partials, n_tiles);
    skipgram_reduce_kernel<<<1, 256, 0, stream>>>(
        partials, (float*)d_out, n_tiles, 1.0f / (float)B);
}